// Block_996432412771
// MI455X (gfx1250) — compile-verified
//
#include <hip/hip_runtime.h>
#include <math.h>

typedef __attribute__((ext_vector_type(2))) float v2f;
typedef __attribute__((ext_vector_type(8))) float v8f;

#define N_EMBD 192
#define N_HEAD 16
#define HEAD   12
#define DPAD   16
#define BATCH  4
#define SEQ    2048
#define M_TOK  (BATCH*SEQ)     /* 8192 */
#define FFN    (4*N_EMBD)      /* 768  */
#define BH     (BATCH*N_HEAD)  /* 64   */

static __device__ __forceinline__ v8f wmma4(v2f a, v2f b, v8f c) {
  // D = A(16x4) * B(4x16) + C(16x16), fp32
  return __builtin_amdgcn_wmma_f32_16x16x4_f32(false, a, false, b, (short)0, c, false, false);
}

// ---------------- RMSNorm: one wave per row (C=192 -> 6 elems/lane) ----------------
__global__ void rmsnorm_kernel(const float* __restrict__ x, const float* __restrict__ w,
                               float* __restrict__ out) {
  int lane = threadIdx.x & 31;
  int wv   = threadIdx.x >> 5;
  int row  = blockIdx.x * 8 + wv;
  const float* xr = x + (size_t)row * N_EMBD;
  float v[6]; float ss = 0.f;
  #pragma unroll
  for (int i = 0; i < 6; i++) { v[i] = xr[lane + 32*i]; ss += v[i]*v[i]; }
  #pragma unroll
  for (int off = 16; off >= 1; off >>= 1) ss += __shfl_xor(ss, off, 32);
  float sc = rsqrtf(ss * (1.0f / N_EMBD) + 1e-5f);
  float* orow = out + (size_t)row * N_EMBD;
  #pragma unroll
  for (int i = 0; i < 6; i++) { int c = lane + 32*i; orow[c] = v[i] * sc * w[c]; }
}

// ---- Pack row-major B[K][N] -> frag-interleaved Bf: (k,n) at ((k>>1)*N + n)*2 + (k&1)
__global__ void pack_b_kernel(const float* __restrict__ B, float* __restrict__ Bf, int N) {
  int idx = blockIdx.x * 256 + threadIdx.x;   // < K*N
  int k = idx / N, n = idx % N;
  Bf[((size_t)(k >> 1) * N + n) * 2 + (k & 1)] = B[idx];
}

// ---- Pack Wq|Wk|Wv [H,C,D] -> frag-interleaved [C x 576] (col = which*192 + h*12 + d)
__global__ void pack_qkv_kernel(const float* __restrict__ Wq, const float* __restrict__ Wk,
                                const float* __restrict__ Wv, float* __restrict__ Bf) {
  int idx = blockIdx.x * 256 + threadIdx.x;   // < 192*576
  int c = idx / 576, j = idx % 576;
  int which = j / 192, r = j % 192;
  int h = r / HEAD, d = r % HEAD;
  const float* W = (which == 0) ? Wq : ((which == 1) ? Wk : Wv);
  Bf[((size_t)(c >> 1) * 576 + j) * 2 + (c & 1)] = W[((size_t)h * N_EMBD + c) * HEAD + d];
}

// ---------------- WMMA GEMM core: one wave -> 16x64 output strip -------------------
// Bf is frag-interleaved: pair (B[k][n],B[k+1][n]) contiguous at ((k>>1)*N + n)*2
__device__ __forceinline__ void gemm16x64(const float* __restrict__ A, const float* __restrict__ Bf,
                                          int K, int lda, int N, int m0, int n0, v8f* acc) {
  int lane = threadIdx.x & 31;
  int mr = lane & 15;
  int kh = (lane >> 4) << 1;
  const float* arow  = A + (size_t)(m0 + mr) * lda;
  const float* bbase = Bf + (size_t)(n0 + mr) * 2;
  #pragma unroll 2
  for (int k = 0; k < K; k += 4) {
    v2f a = *(const v2f*)(arow + k + kh);
    const float* bk = bbase + (size_t)((k + kh) >> 1) * (2 * N);
    #pragma unroll
    for (int t = 0; t < 4; t++) {
      v2f b = *(const v2f*)(bk + t * 32);
      acc[t] = wmma4(a, b, acc[t]);
    }
  }
}

// ---------------- QKV GEMM: [8192x192]*[192x576] -> scatter to padded Q/K/V --------
// Qp,Kp: [BH][T][16] row-major.  Vp: s-pair interleaved: (s,d) at (s>>1)*32 + d*2 + (s&1)
__global__ void gemm_qkv_kernel(const float* __restrict__ A, const float* __restrict__ Bf,
                                float* __restrict__ Qp, float* __restrict__ Kp,
                                float* __restrict__ Vp) {
  int wv = threadIdx.x >> 5;
  int tile = blockIdx.x * 8 + wv;
  const int TN = 576 / 64;                 // 9
  int m0 = (tile / TN) * 16;
  int n0 = (tile % TN) * 64;
  v8f acc[4] = {{}, {}, {}, {}};
  gemm16x64(A, Bf, N_EMBD, N_EMBD, 576, m0, n0, acc);
  int lane = threadIdx.x & 31;
  int mr = lane & 15, half = lane >> 4;
  #pragma unroll
  for (int t = 0; t < 4; t++) {
    int j = n0 + t * 16 + mr;
    int which = j / 192, r = j % 192, h = r / HEAD, d = r % HEAD;
    #pragma unroll
    for (int v = 0; v < 8; v++) {
      int m = m0 + v + 8 * half;
      int b = m >> 11, tq = m & 2047;
      size_t hb = ((size_t)(b * N_HEAD + h)) * SEQ * DPAD;
      if (which == 0)      Qp[hb + (size_t)tq * DPAD + d] = acc[t][v];
      else if (which == 1) Kp[hb + (size_t)tq * DPAD + d] = acc[t][v];
      else                 Vp[hb + (size_t)(tq >> 1) * 32 + d * 2 + (tq & 1)] = acc[t][v];
    }
  }
}

// ---------------- Flash attention: one wave per 16-query tile ----------------------
__global__ void attn_kernel(const float* __restrict__ Qp, const float* __restrict__ Kp,
                            const float* __restrict__ Vp, float* __restrict__ sa) {
  __shared__ float Pb[4][256];
  int lane = threadIdx.x & 31, wv = threadIdx.x >> 5;
  int mr = lane & 15, half = lane >> 4, kh = half << 1;
  int qt = wv * 32 + blockIdx.x;           // strided for load balance, qt in [0,128)
  int bh = blockIdx.y;
  const size_t base = (size_t)bh * SEQ * DPAD;
  const float* Qb = Qp + base + (size_t)qt * 16 * DPAD;

  v2f qf[4];
  #pragma unroll
  for (int ck = 0; ck < 4; ck++) qf[ck] = *(const v2f*)(Qb + mr * DPAD + ck * 4 + kh);

  v8f o = {};
  float mrun[8], lrun[8];
  #pragma unroll
  for (int v = 0; v < 8; v++) { mrun[v] = -INFINITY; lrun[v] = 0.f; }
  const float scale = 0.28867513459481287f;   // 1/sqrt(12)

  for (int kt = 0; kt <= qt; ++kt) {
    const float* Kt = Kp + base + (size_t)kt * 256;
    const float* Vt = Vp + base + (size_t)kt * 256;
    if (kt < qt) {                           // prefetch next tiles -> global_prefetch_b8
      __builtin_prefetch(Kt + 256, 0, 0);
      __builtin_prefetch(Vt + 256, 0, 0);
    }
    v8f s = {};
    #pragma unroll
    for (int ck = 0; ck < 4; ck++) {
      v2f bf = *(const v2f*)(Kt + mr * DPAD + ck * 4 + kh);  // K[key=mr][d], contiguous
      s = wmma4(qf[ck], bf, s);
    }
    #pragma unroll
    for (int v = 0; v < 8; v++) {
      float sv = s[v] * scale;
      sv = (kt == qt && mr > v + 8 * half) ? -INFINITY : sv;  // causal mask on diagonal
      float rmax = sv;
      #pragma unroll
      for (int off = 8; off >= 1; off >>= 1) rmax = fmaxf(rmax, __shfl_xor(rmax, off, 32));
      float mnew  = fmaxf(mrun[v], rmax);
      float alpha = __expf(mrun[v] - mnew);
      float pe    = __expf(sv - mnew);
      float rsum  = pe;
      #pragma unroll
      for (int off = 8; off >= 1; off >>= 1) rsum += __shfl_xor(rsum, off, 32);
      lrun[v] = lrun[v] * alpha + rsum;
      mrun[v] = mnew;
      o[v]    = o[v] * alpha;
      Pb[wv][(v + 8 * half) * 16 + mr] = pe;   // C/D layout -> row-major P tile in LDS
    }
    __asm__ volatile("" ::: "memory");          // LDS ops are in-order within a wave
    #pragma unroll
    for (int ck = 0; ck < 4; ck++) {
      v2f a = *(const v2f*)(&Pb[wv][mr * 16 + ck * 4 + kh]); // P[q=mr][s], A-frag layout
      int s0 = ck * 4 + kh;
      v2f bf = *(const v2f*)(Vt + (s0 >> 1) * 32 + mr * 2);  // (V[s0][d],V[s0+1][d]) pair
      o = wmma4(a, bf, o);
    }
  }

  int b = bh >> 4, h = bh & 15;
  if (mr < HEAD) {
    #pragma unroll
    for (int v = 0; v < 8; v++) {
      int q = qt * 16 + v + 8 * half;
      sa[((size_t)(b * SEQ + q)) * N_EMBD + h * HEAD + mr] = o[v] / lrun[v];
    }
  }
}

// ---------------- Proj GEMM + bias + residual -------------------------------------
__global__ void gemm_proj_kernel(const float* __restrict__ A, const float* __restrict__ Bf,
                                 const float* __restrict__ bias, const float* __restrict__ resid,
                                 float* __restrict__ out) {
  int wv = threadIdx.x >> 5;
  int tile = blockIdx.x * 8 + wv;
  const int TN = 192 / 64;                 // 3
  int m0 = (tile / TN) * 16, n0 = (tile % TN) * 64;
  v8f acc[4] = {{}, {}, {}, {}};
  gemm16x64(A, Bf, 192, 192, 192, m0, n0, acc);
  int lane = threadIdx.x & 31; int mr = lane & 15, half = lane >> 4;
  #pragma unroll
  for (int t = 0; t < 4; t++) {
    int j = n0 + t * 16 + mr;
    float bj = bias[j];
    #pragma unroll
    for (int v = 0; v < 8; v++) {
      int m = m0 + v + 8 * half;
      size_t ofs = (size_t)m * 192 + j;
      out[ofs] = acc[t][v] + bj + resid[ofs];
    }
  }
}

// ---------------- FF1 GEMM + SiLU --------------------------------------------------
__global__ void gemm_ff1_kernel(const float* __restrict__ A, const float* __restrict__ Bf,
                                float* __restrict__ out) {
  int wv = threadIdx.x >> 5;
  int tile = blockIdx.x * 8 + wv;
  const int TN = FFN / 64;                 // 12
  int m0 = (tile / TN) * 16, n0 = (tile % TN) * 64;
  v8f acc[4] = {{}, {}, {}, {}};
  gemm16x64(A, Bf, 192, 192, FFN, m0, n0, acc);
  int lane = threadIdx.x & 31; int mr = lane & 15, half = lane >> 4;
  #pragma unroll
  for (int t = 0; t < 4; t++) {
    int j = n0 + t * 16 + mr;
    #pragma unroll
    for (int v = 0; v < 8; v++) {
      int m = m0 + v + 8 * half;
      float val = acc[t][v];
      out[(size_t)m * FFN + j] = val / (1.0f + __expf(-val));   // silu
    }
  }
}

// ---------------- FF2 GEMM + residual -> d_out -------------------------------------
__global__ void gemm_ff2_kernel(const float* __restrict__ A, const float* __restrict__ Bf,
                                const float* __restrict__ resid, float* __restrict__ out) {
  int wv = threadIdx.x >> 5;
  int tile = blockIdx.x * 8 + wv;
  const int TN = 192 / 64;                 // 3
  int m0 = (tile / TN) * 16, n0 = (tile % TN) * 64;
  v8f acc[4] = {{}, {}, {}, {}};
  gemm16x64(A, Bf, FFN, FFN, 192, m0, n0, acc);
  int lane = threadIdx.x & 31; int mr = lane & 15, half = lane >> 4;
  #pragma unroll
  for (int t = 0; t < 4; t++) {
    int j = n0 + t * 16 + mr;
    #pragma unroll
    for (int v = 0; v < 8; v++) {
      int m = m0 + v + 8 * half;
      size_t ofs = (size_t)m * 192 + j;
      out[ofs] = acc[t][v] + resid[ofs];
    }
  }
}

extern "C" void kernel_launch(void* const* d_in, const int* in_sizes, int n_in,
                              void* d_out, int out_size, void* d_ws, size_t ws_size,
                              hipStream_t stream) {
  (void)in_sizes; (void)n_in; (void)out_size; (void)ws_size;
  const float* x     = (const float*)d_in[0];
  const float* ln1_w = (const float*)d_in[1];
  const float* ln2_w = (const float*)d_in[2];
  const float* Wq    = (const float*)d_in[3];
  const float* Wk    = (const float*)d_in[4];
  const float* Wv    = (const float*)d_in[5];
  const float* Wproj = (const float*)d_in[6];
  const float* bproj = (const float*)d_in[7];
  const float* W1    = (const float*)d_in[8];
  const float* W2    = (const float*)d_in[9];
  float* ws = (float*)d_ws;

  size_t off = 0;
  float* h    = ws + off; off += (size_t)M_TOK * N_EMBD;   // rmsnorm1 out
  float* wqkv = ws + off; off += (size_t)192 * 576;        // packed QKV weights
  float* wpjf = ws + off; off += (size_t)192 * 192;        // packed Wproj
  float* w1f  = ws + off; off += (size_t)192 * FFN;        // packed W1
  float* w2f  = ws + off; off += (size_t)FFN * 192;        // packed W2
  float* Qp   = ws + off; off += (size_t)BH * SEQ * DPAD;  // padded Q
  float* Kp   = ws + off; off += (size_t)BH * SEQ * DPAD;  // padded K
  float* Vp   = ws + off; off += (size_t)BH * SEQ * DPAD;  // padded V (s-interleaved)
  float* sa   = ws + off; off += (size_t)M_TOK * N_EMBD;   // attention out
  float* x1   = ws + off; off += (size_t)M_TOK * N_EMBD;   // residual stream
  float* h2   = ws + off; off += (size_t)M_TOK * N_EMBD;   // rmsnorm2 out
  float* u    = ws + off; off += (size_t)M_TOK * FFN;      // silu(h2@W1)
  float* out  = (float*)d_out;

  // zero padded QKV (pads d=12..15 must be 0 for the d-reduction)
  hipMemsetAsync(Qp, 0, (size_t)3 * BH * SEQ * DPAD * sizeof(float), stream);

  pack_qkv_kernel<<<(192 * 576) / 256, 256, 0, stream>>>(Wq, Wk, Wv, wqkv);
  pack_b_kernel<<<(192 * 192) / 256, 256, 0, stream>>>(Wproj, wpjf, 192);
  pack_b_kernel<<<(192 * FFN) / 256, 256, 0, stream>>>(W1, w1f, FFN);
  pack_b_kernel<<<(FFN * 192) / 256, 256, 0, stream>>>(W2, w2f, 192);

  rmsnorm_kernel<<<M_TOK / 8, 256, 0, stream>>>(x, ln1_w, h);
  gemm_qkv_kernel<<<(512 * 9) / 8, 256, 0, stream>>>(h, wqkv, Qp, Kp, Vp);
  attn_kernel<<<dim3(32, BH), 128, 0, stream>>>(Qp, Kp, Vp, sa);
  gemm_proj_kernel<<<(512 * 3) / 8, 256, 0, stream>>>(sa, wpjf, bproj, x, x1);
  rmsnorm_kernel<<<M_TOK / 8, 256, 0, stream>>>(x1, ln2_w, h2);
  gemm_ff1_kernel<<<(512 * 12) / 8, 256, 0, stream>>>(h2, w1f, u);
  gemm_ff2_kernel<<<(512 * 3) / 8, 256, 0, stream>>>(u, w2f, x1, out);
}